// FlexiConv_90941637526060
// MI455X (gfx1250) — compile-verified
//
#include <hip/hip_runtime.h>

typedef __attribute__((ext_vector_type(2))) float v2f;
typedef __attribute__((ext_vector_type(8))) float v8f;

#define B_    16
#define CIN   64
#define COUT  128
#define H_    224
#define W_    224
#define TY    4      // pixel rows per block
#define TX    16     // pixel cols per block (= WMMA N)
#define XS_R  (TY + 2)
#define XS_C  (TX + 2)

#define HAS_ASYNC_LDS __has_builtin(__builtin_amdgcn_global_load_async_to_lds_b32)

typedef __attribute__((address_space(1))) int  g_int;
typedef __attribute__((address_space(3))) int  l_int;

// Fused depthwise(3x3,pad1)+bias+channel-scale -> pointwise(1x1,64->128)+bias.
// GEMM mapping per wave:  D[mo=outch 16][p=pixel 16] = sum_c pw_w[mo][c] * weighted[c][p]
// done as 16 chained V_WMMA_F32_16X16X4_F32 (K=64).
// Input halo staging uses GLOBAL_LOAD_ASYNC_TO_LDS_B32 (ASYNCcnt) when available.
__global__ __launch_bounds__(256)
void fused_dwsep_wmma_f32(const float* __restrict__ x,
                          const float* __restrict__ dw_w,
                          const float* __restrict__ dw_b,
                          const float* __restrict__ pw_w,
                          const float* __restrict__ pw_b,
                          const float* __restrict__ gw,
                          float* __restrict__ out)
{
    __shared__ float sX[CIN][XS_R][XS_C];   // input patch + halo (27648 B)
    __shared__ float sA[TY][CIN][TX];       // weighted activations  (16384 B)
    __shared__ float sScale[CIN];
    __shared__ float sDwW[CIN][9];
    __shared__ float sDwB[CIN];

    const int tid  = threadIdx.x;
    const int lane = tid & 31;
    const int wave = tid >> 5;              // 0..7 -> out-channel tile

    const int x0 = blockIdx.x * TX;
    const int y0 = blockIdx.y * TY;
    const int b  = blockIdx.z;
    const int n0 = wave * 16;               // out-channel tile base

    // ---- pw_w fragments: WMMA A-matrix (M=outch x K=4), 16 chunks => K=64 ----
    // lane l: mo = l%16 ; holds K = 4k + (l<16 ? {0,1} : {2,3})
    v2f afrag[16];
    {
        const int mo = lane & 15;
        const int kb = (lane < 16) ? 0 : 2;
        const float* wrow = pw_w + (n0 + mo) * CIN;   // pw_w is [COUT][CIN]
        #pragma unroll
        for (int k = 0; k < 16; ++k) {
            afrag[k].x = wrow[4 * k + kb + 0];
            afrag[k].y = wrow[4 * k + kb + 1];
        }
    }
    // ---- output bias per lane: channel = n0 + j + (lane<16 ? 0 : 8) ----
    float bout[8];
    {
        const int half = (lane < 16) ? 0 : 8;
        #pragma unroll
        for (int j = 0; j < 8; ++j) bout[j] = pw_b[n0 + half + j];
    }

    // ---- depthwise params + gaussian scale into LDS ----
    for (int i = tid; i < CIN * 9; i += 256) sDwW[i / 9][i % 9] = dw_w[i];
    if (tid < CIN) {
        float s = 0.f;
        #pragma unroll
        for (int j = 0; j < 9; ++j) { float g = gw[tid * 9 + j]; s += __expf(-g * g); }
        sScale[tid] = s;
        sDwB[tid]   = dw_b[tid];
    }

    // ---- stage input patch (with zero halo) ----
#if HAS_ASYNC_LDS
    for (int i = tid; i < CIN * XS_R * XS_C; i += 256) {
        const int c  = i / (XS_R * XS_C);
        const int rr = (i % (XS_R * XS_C)) / XS_C;
        const int cc = i % XS_C;
        const int gy = y0 + rr - 1;
        const int gx = x0 + cc - 1;
        float* lp = &sX[c][rr][cc];
        if (gy >= 0 && gy < H_ && gx >= 0 && gx < W_) {
            const float* gp = &x[((b * CIN + c) * H_ + gy) * W_ + gx];
            __builtin_amdgcn_global_load_async_to_lds_b32(
                (g_int*)(uintptr_t)gp,
                (l_int*)(uint32_t)(uintptr_t)lp,
                0, 0);
        } else {
            *lp = 0.f;   // zero halo (disjoint addresses from async path)
        }
    }
  #if __has_builtin(__builtin_amdgcn_s_wait_asynccnt)
    __builtin_amdgcn_s_wait_asynccnt(0);
  #else
    asm volatile("s_wait_asynccnt 0x0" ::: "memory");
  #endif
#else
    for (int i = tid; i < CIN * XS_R * XS_C; i += 256) {
        const int c  = i / (XS_R * XS_C);
        const int rr = (i % (XS_R * XS_C)) / XS_C;
        const int cc = i % XS_C;
        const int gy = y0 + rr - 1;
        const int gx = x0 + cc - 1;
        float v = 0.f;
        if (gy >= 0 && gy < H_ && gx >= 0 && gx < W_)
            v = x[((b * CIN + c) * H_ + gy) * W_ + gx];
        sX[c][rr][cc] = v;
    }
#endif
    __syncthreads();

    // ---- depthwise + bias + scale -> sA[r][c][m]  (each thread: 4 ch x 4 rows) ----
    {
        const int m  = tid & 15;
        const int cg = tid >> 4;            // 0..15
        #pragma unroll
        for (int ci = 0; ci < 4; ++ci) {
            const int c = cg * 4 + ci;
            const float sc   = sScale[c];
            const float bias = sDwB[c];
            const float* wk  = &sDwW[c][0];
            #pragma unroll
            for (int r = 0; r < TY; ++r) {
                float acc = bias;
                #pragma unroll
                for (int dy = 0; dy < 3; ++dy)
                    #pragma unroll
                    for (int dx = 0; dx < 3; ++dx)
                        acc += sX[c][r + dy][m + dx] * wk[dy * 3 + dx];
                sA[r][c][m] = acc * sc;
            }
        }
    }
    __syncthreads();

    // ---- WMMA phase: per row tile, K=64 via 16 x wmma_f32_16x16x4_f32 ----
    const int p  = lane & 15;               // pixel (N index)
    const int kb = (lane < 16) ? 0 : 2;     // K sub-offset within chunk
    #pragma unroll
    for (int r = 0; r < TY; ++r) {
        v8f acc = {0.f, 0.f, 0.f, 0.f, 0.f, 0.f, 0.f, 0.f};
        #pragma unroll
        for (int k = 0; k < 16; ++k) {
            v2f bfrag;
            bfrag.x = sA[r][4 * k + kb + 0][p];
            bfrag.y = sA[r][4 * k + kb + 1][p];
            acc = __builtin_amdgcn_wmma_f32_16x16x4_f32(
                /*neg_a=*/false, afrag[k],
                /*neg_b=*/false, bfrag,
                /*c_mod=*/(short)0, acc,
                /*reuse_a=*/false, /*reuse_b=*/false);
        }
        const int gy   = y0 + r;
        const int half = (lane < 16) ? 0 : 8;
        #pragma unroll
        for (int j = 0; j < 8; ++j) {
            const int nch = n0 + half + j;  // D row M = j + half
            out[((size_t)(b * COUT + nch) * H_ + gy) * W_ + (x0 + p)] = acc[j] + bout[j];
        }
    }
}

extern "C" void kernel_launch(void* const* d_in, const int* in_sizes, int n_in,
                              void* d_out, int out_size, void* d_ws, size_t ws_size,
                              hipStream_t stream) {
    (void)in_sizes; (void)n_in; (void)d_ws; (void)ws_size; (void)out_size;
    const float* x    = (const float*)d_in[0];
    const float* dw_w = (const float*)d_in[1];
    const float* dw_b = (const float*)d_in[2];
    const float* pw_w = (const float*)d_in[3];
    const float* pw_b = (const float*)d_in[4];
    const float* gw   = (const float*)d_in[5];
    float* out = (float*)d_out;

    dim3 grid(W_ / TX, H_ / TY, B_);   // 14 x 56 x 16
    dim3 block(256);
    fused_dwsep_wmma_f32<<<grid, block, 0, stream>>>(x, dw_w, dw_b, pw_w, pw_b, gw, out);
}